// Monotonic_Attention_61203283968867
// MI455X (gfx1250) — compile-verified
//
#include <hip/hip_runtime.h>
#include <math.h>

// Problem sizes (fixed by the reference)
#define BQ 64
#define SQ 4096
#define DQ 256
#define UQ 256

typedef __attribute__((ext_vector_type(2))) float v2f;
typedef __attribute__((ext_vector_type(8))) float v8f;

// ---------------------------------------------------------------------------
// Kernel 1: V = dec_state @ W1^T  (M=B=64, N=D=256, K=U=256), f32 WMMA.
// V[b,d] = sum_u dec[b,u] * W1[d,u].
// One wave (32 threads) computes one 16x16 tile, K-loop in steps of 4 using
// v_wmma_f32_16x16x4_f32. Grid = (64/16)*(256/16) = 64 tiles.
// ---------------------------------------------------------------------------
__global__ __launch_bounds__(32) void wmma_v_kernel(const float* __restrict__ dec,
                                                    const float* __restrict__ W1,
                                                    float* __restrict__ V) {
  const int tile = blockIdx.x;
  const int m0 = (tile >> 4) << 4;   // 4 tiles over B
  const int n0 = (tile & 15) << 4;   // 16 tiles over D
  const int lane = threadIdx.x;
  const int half = lane >> 4;        // 0: lanes 0-15, 1: lanes 16-31
  const int lr   = lane & 15;

  v8f acc = {};
  for (int k0 = 0; k0 < UQ; k0 += 4) {
    const int k = k0 + 2 * half;
    // A 16x4 (MxK): lane half selects K pair, lr = M row
    const float* ap = dec + (m0 + lr) * UQ + k;
    v2f a; a.x = ap[0]; a.y = ap[1];
    // B 4x16 (KxN): B[k][n] = W1[n][k]; lr = N col, half selects K pair
    const float* bp = W1 + (n0 + lr) * UQ + k;
    v2f b; b.x = bp[0]; b.y = bp[1];
    acc = __builtin_amdgcn_wmma_f32_16x16x4_f32(
        /*neg_a=*/false, a, /*neg_b=*/false, b,
        /*c_mod=*/(short)0, acc, /*reuse_a=*/false, /*reuse_b=*/false);
  }
  // C/D 16x16 f32 layout: VGPR r -> M = r (lanes 0-15) or 8+r (lanes 16-31)
#pragma unroll
  for (int r = 0; r < 8; ++r) {
    V[(m0 + r + 8 * half) * DQ + n0 + lr] = acc[r];
  }
}

// ---------------------------------------------------------------------------
// Kernel 2: cb[b] = b1 . dec[b,:]  (tiny)
// ---------------------------------------------------------------------------
__global__ __launch_bounds__(32) void cb_kernel(const float* __restrict__ dec,
                                                const float* __restrict__ b1,
                                                float* __restrict__ cb) {
  const int b = blockIdx.x;
  const int l = threadIdx.x;
  float s = 0.0f;
  for (int u = l; u < UQ; u += 32) s += b1[u] * dec[b * UQ + u];
#pragma unroll
  for (int off = 16; off; off >>= 1) s += __shfl_xor(s, off, 32);
  if (l == 0) cb[b] = s;
}

// ---------------------------------------------------------------------------
// Kernel 3: p_pre[b,s] = sigmoid(enc[b,s,:] . V[b,:] + cb[b])
// HBM-bound streaming pass over enc (256 MB). One wave per s-row, float4
// loads (512B/wave per load), butterfly reduce over wave32.
// ---------------------------------------------------------------------------
__global__ __launch_bounds__(256) void score_kernel(const float* __restrict__ enc,
                                                    const float* __restrict__ V,
                                                    const float* __restrict__ cb,
                                                    float* __restrict__ p_pre) {
  __shared__ float vsh[DQ];
  __shared__ float cbs;
  const int b   = blockIdx.x;
  const int s0  = blockIdx.y * 256;
  const int tid = threadIdx.x;
  vsh[tid] = V[b * DQ + tid];
  if (tid == 0) cbs = cb[b];
  __syncthreads();

  const int wave = tid >> 5, lane = tid & 31;
  const float4 v0 = *(const float4*)(vsh + lane * 4);
  const float4 v1 = *(const float4*)(vsh + 128 + lane * 4);

  for (int row = wave; row < 256; row += 8) {
    const int s = s0 + row;
    const float4* e = (const float4*)(enc + ((size_t)b * SQ + s) * DQ);
    const float4 e0 = e[lane];
    const float4 e1 = e[lane + 32];
    float acc = e0.x * v0.x + e0.y * v0.y + e0.z * v0.z + e0.w * v0.w +
                e1.x * v1.x + e1.y * v1.y + e1.z * v1.z + e1.w * v1.w;
#pragma unroll
    for (int off = 16; off; off >>= 1) acc += __shfl_xor(acc, off, 32);
    if (lane == 0) {
      const float sc = acc + cbs;
      p_pre[(size_t)b * SQ + s] = 1.0f / (1.0f + __expf(-sc));
    }
  }
}

// ---------------------------------------------------------------------------
// Kernel 4: per-batch scans. cumsum(prev_att) -> p = p_pre * cumsum ->
// attention = p * exclusive_cumprod(1-p). One 1024-thread block per batch,
// 4 elements/thread, Hillis-Steele LDS scans (add then mul).
// ---------------------------------------------------------------------------
__global__ __launch_bounds__(1024) void scan_kernel(const float* __restrict__ prev_att,
                                                    const float* __restrict__ p_pre,
                                                    float* __restrict__ att_out) {
  __shared__ float sc[1024];
  const int b = blockIdx.x;
  const int t = threadIdx.x;
  const size_t base = (size_t)b * SQ + (size_t)t * 4;

  const float a0 = prev_att[base + 0], a1 = prev_att[base + 1];
  const float a2 = prev_att[base + 2], a3 = prev_att[base + 3];
  const float l0 = a0, l1 = l0 + a1, l2 = l1 + a2, l3 = l2 + a3;

  // inclusive add-scan of per-thread totals
  float x = l3;
  sc[t] = x;
  __syncthreads();
  for (int off = 1; off < 1024; off <<= 1) {
    const float y = (t >= off) ? sc[t - off] : 0.0f;
    __syncthreads();
    x += y;
    sc[t] = x;
    __syncthreads();
  }
  const float ex = (t > 0) ? sc[t - 1] : 0.0f;
  const float c0 = ex + l0, c1 = ex + l1, c2 = ex + l2, c3 = ex + l3;

  const float p0 = p_pre[base + 0] * c0, p1 = p_pre[base + 1] * c1;
  const float p2 = p_pre[base + 2] * c2, p3 = p_pre[base + 3] * c3;
  const float q0 = 1.0f - p0, q1 = 1.0f - p1, q2 = 1.0f - p2, q3 = 1.0f - p3;

  __syncthreads();  // done reading sc from the add-scan

  // inclusive mul-scan of per-thread products
  float m = q0 * q1 * q2 * q3;
  sc[t] = m;
  __syncthreads();
  for (int off = 1; off < 1024; off <<= 1) {
    const float y = (t >= off) ? sc[t - off] : 1.0f;
    __syncthreads();
    m *= y;
    sc[t] = m;
    __syncthreads();
  }
  const float E0 = (t > 0) ? sc[t - 1] : 1.0f;  // exclusive prefix product
  const float E1 = E0 * q0, E2 = E1 * q1, E3 = E2 * q2;

  att_out[base + 0] = p0 * E0;
  att_out[base + 1] = p1 * E1;
  att_out[base + 2] = p2 * E2;
  att_out[base + 3] = p3 * E3;
}

// ---------------------------------------------------------------------------
// Kernel 5: partial context sums (second streaming pass over enc, 256 MB).
// Grid (B, 16): block handles 256 s-rows; thread t owns output dim d=t.
// Coalesced 1KB row loads; deterministic partials (no FP atomics).
// ---------------------------------------------------------------------------
__global__ __launch_bounds__(256) void ctx_partial_kernel(const float* __restrict__ enc,
                                                          const float* __restrict__ att,
                                                          float* __restrict__ part) {
  __shared__ float ash[256];
  const int b  = blockIdx.x;
  const int c  = blockIdx.y;
  const int t  = threadIdx.x;
  const int s0 = c * 256;
  ash[t] = att[(size_t)b * SQ + s0 + t];
  __syncthreads();

  float acc = 0.0f;
  const float* ep = enc + ((size_t)b * SQ + s0) * DQ + t;
#pragma unroll 4
  for (int i = 0; i < 256; ++i) acc += ash[i] * ep[(size_t)i * DQ];
  part[((size_t)c * BQ + b) * DQ + t] = acc;
}

// ---------------------------------------------------------------------------
// Kernel 6: reduce the 16 partials in fixed order -> context [B, D]
// ---------------------------------------------------------------------------
__global__ __launch_bounds__(256) void ctx_reduce_kernel(const float* __restrict__ part,
                                                         float* __restrict__ ctx) {
  const int b = blockIdx.x;
  const int t = threadIdx.x;
  float s = 0.0f;
#pragma unroll
  for (int c = 0; c < 16; ++c) s += part[((size_t)c * BQ + b) * DQ + t];
  ctx[b * DQ + t] = s;
}

// ---------------------------------------------------------------------------
extern "C" void kernel_launch(void* const* d_in, const int* in_sizes, int n_in,
                              void* d_out, int out_size, void* d_ws, size_t ws_size,
                              hipStream_t stream) {
  const float* enc  = (const float*)d_in[0];  // [B,S,D]
  const float* dec  = (const float*)d_in[1];  // [B,U]
  const float* prev = (const float*)d_in[2];  // [B,S,1]
  const float* W1   = (const float*)d_in[3];  // [D,U]
  const float* b1   = (const float*)d_in[4];  // [U]

  float* out = (float*)d_out;
  float* ctx = out;             // context_vec [B,D] first
  float* att = out + BQ * DQ;   // attention [B,S,1] second

  float* ws    = (float*)d_ws;
  float* V     = ws;                              // B*D  = 16384 floats
  float* cb    = V + BQ * DQ;                     // B    = 64
  float* p_pre = cb + BQ;                         // B*S  = 262144
  float* part  = p_pre + (size_t)BQ * SQ;         // 16*B*D = 262144

  wmma_v_kernel<<<dim3(64), dim3(32), 0, stream>>>(dec, W1, V);
  cb_kernel<<<dim3(BQ), dim3(32), 0, stream>>>(dec, b1, cb);
  score_kernel<<<dim3(BQ, SQ / 256), dim3(256), 0, stream>>>(enc, V, cb, p_pre);
  scan_kernel<<<dim3(BQ), dim3(1024), 0, stream>>>(prev, p_pre, att);
  ctx_partial_kernel<<<dim3(BQ, SQ / 256), dim3(256), 0, stream>>>(enc, att, part);
  ctx_reduce_kernel<<<dim3(BQ), dim3(256), 0, stream>>>(part, ctx);
}